// NormalLoss_50173807952230
// MI455X (gfx1250) — compile-verified
//
#include <hip/hip_runtime.h>
#include <hip/hip_bf16.h>

// ---------------------------------------------------------------------------
// NormalLoss on MI455X (gfx1250, wave32).
// Pipeline:
//   1) gather_zero:   pred_g = pred[idx12], zero d_out[0]
//   2) knn_normals:   per (batch, 128-row block): cloud -> LDS, Gram tiles via
//                     V_WMMA_F32_16X16X4_F32 (K=3 padded to 4), per-row top-10
//                     selection, closed-form 3x3 PCA normal.   (run for gt, pred_g)
//   3) loss_reduce:   1 - cosine, sum over points, mean over batch -> d_out[0]
// ---------------------------------------------------------------------------

typedef __attribute__((ext_vector_type(2))) float v2f;
typedef __attribute__((ext_vector_type(8))) float v8f;

#define BATCH 16
#define NPTS  2048
#define KNN   10

// ---------------------------------------------------------------------------
// Smallest eigenvector of symmetric 3x3 [[a00,a01,a02],[a01,a11,a12],[a02,a12,a22]]
// Trigonometric eigenvalues + cross-product eigenvector (robust pair choice).
// ---------------------------------------------------------------------------
__device__ __forceinline__ void smallest_evec(float a00, float a01, float a02,
                                              float a11, float a12, float a22,
                                              float& ex, float& ey, float& ez) {
    float p1 = a01 * a01 + a02 * a02 + a12 * a12;
    float q  = (a00 + a11 + a22) * (1.0f / 3.0f);
    float b00 = a00 - q, b11 = a11 - q, b22 = a22 - q;
    float p2 = b00 * b00 + b11 * b11 + b22 * b22 + 2.0f * p1;
    float p  = sqrtf(p2 * (1.0f / 6.0f) + 1e-30f);
    float ip = 1.0f / p;
    float c00 = b00 * ip, c01 = a01 * ip, c02 = a02 * ip;
    float c11 = b11 * ip, c12 = a12 * ip, c22 = b22 * ip;
    float detB = c00 * (c11 * c22 - c12 * c12)
               - c01 * (c01 * c22 - c12 * c02)
               + c02 * (c01 * c12 - c11 * c02);
    float r = fminf(1.0f, fmaxf(-1.0f, 0.5f * detB));
    float phi = acosf(r) * (1.0f / 3.0f);
    // smallest eigenvalue: q + 2p*cos(phi + 2*pi/3)
    float lam = q + 2.0f * p * cosf(phi + 2.0943951023931953f);

    float m00 = a00 - lam, m11 = a11 - lam, m22 = a22 - lam;
    // cross products of row pairs of (A - lam I)
    float v0x = a01 * a12 - a02 * m11;
    float v0y = a02 * a01 - m00 * a12;
    float v0z = m00 * m11 - a01 * a01;
    float v1x = a01 * m22 - a02 * a12;
    float v1y = a02 * a02 - m00 * m22;
    float v1z = m00 * a12 - a01 * a02;
    float v2x = m11 * m22 - a12 * a12;
    float v2y = a12 * a02 - a01 * m22;
    float v2z = a01 * a12 - m11 * a02;
    float n0 = v0x * v0x + v0y * v0y + v0z * v0z;
    float n1 = v1x * v1x + v1y * v1y + v1z * v1z;
    float n2 = v2x * v2x + v2y * v2y + v2z * v2z;
    float vx = v0x, vy = v0y, vz = v0z, nn = n0;
    if (n1 > nn) { vx = v1x; vy = v1y; vz = v1z; nn = n1; }
    if (n2 > nn) { vx = v2x; vy = v2y; vz = v2z; nn = n2; }
    if (nn < 1e-30f) { vx = 0.0f; vy = 0.0f; vz = 1.0f; nn = 1.0f; }
    float inv = rsqrtf(nn);
    ex = vx * inv; ey = vy * inv; ez = vz * inv;
    // deterministic sign (eigh's sign is convention-dependent anyway)
    if (ez < 0.0f || (ez == 0.0f && (ey < 0.0f || (ey == 0.0f && ex < 0.0f)))) {
        ex = -ex; ey = -ey; ez = -ez;
    }
}

// ---------------------------------------------------------------------------
// Kernel 1: pred_g = pred[b, idx12[b,n], :]; thread (0,0) zeroes the output.
// ---------------------------------------------------------------------------
__global__ void gather_zero(const float* __restrict__ pred,
                            const int* __restrict__ idx12,
                            float* __restrict__ predg,
                            float* __restrict__ out) {
    if (blockIdx.x == 0 && threadIdx.x == 0) out[0] = 0.0f;
    int t = blockIdx.x * blockDim.x + threadIdx.x;
    if (t < BATCH * NPTS) {
        int b = t / NPTS;
        int s = idx12[t];
        const float* src = pred + ((size_t)b * NPTS + s) * 3;
        float* dst = predg + (size_t)t * 3;
        dst[0] = src[0]; dst[1] = src[1]; dst[2] = src[2];
    }
}

// ---------------------------------------------------------------------------
// Kernel 2: kNN (k=10) + PCA normals.
//  grid  = (NPTS/128, BATCH), block = 128 (4 waves); each wave owns 32 rows.
//  Whole batch cloud lives in LDS (SoA + |p|^2): 32KB; per-wave 32x17 Gram
//  transpose slab: 8.5KB.  Gram tiles via V_WMMA_F32_16X16X4_F32.
//
//  16x4 f32 A fragment (ISA 7.12.2): lanes 0-15 hold (K0,K1)=(x,y) of row M=lane,
//  lanes 16-31 hold (K2,K3)=(z,0) of row M=lane-16.  B (4x16) built with the
//  mirrored layout over the 16 column points, so A and B fragments are
//  constructed identically from the point cloud.
// ---------------------------------------------------------------------------
__global__ void __launch_bounds__(128)
knn_normals_kernel(const float* __restrict__ pts, float* __restrict__ normals) {
    __shared__ float xs[NPTS], ys[NPTS], zs[NPTS], sqv[NPTS];
    __shared__ float tile[4][32 * 17];

    const int b   = blockIdx.y;
    const int tid = threadIdx.x;
    const float* P = pts + (size_t)b * NPTS * 3;

    for (int n = tid; n < NPTS; n += 128) {
        float x = P[n * 3 + 0], y = P[n * 3 + 1], z = P[n * 3 + 2];
        xs[n] = x; ys[n] = y; zs[n] = z;
        sqv[n] = x * x + y * y + z * z;
    }
    __syncthreads();

    const int wave  = tid >> 5;
    const int lane  = tid & 31;
    const int rBase = blockIdx.x * 128 + wave * 32;
    float* tbuf = tile[wave];

    const bool lo  = lane < 16;
    const int  sub = lane & 15;

    // A fragments: rows rBase..rBase+15 (a0) and rBase+16..rBase+31 (a1)
    v2f a0, a1;
    {
        int r  = rBase + sub;
        int r2 = r + 16;
        a0.x = lo ? xs[r]  : zs[r];   a0.y = lo ? ys[r]  : 0.0f;
        a1.x = lo ? xs[r2] : zs[r2];  a1.y = lo ? ys[r2] : 0.0f;
    }
    const float sqR = sqv[rBase + lane];

    float nd[KNN]; int ni[KNN];
#pragma unroll
    for (int j = 0; j < KNN; ++j) { nd[j] = 3.4e38f; ni[j] = 0; }

    const int Mbase = (lane >> 4) << 3;  // 0 or 8
    const int Ncol  = sub;

    for (int c0 = 0; c0 < NPTS; c0 += 16) {
        // B fragment for 16 column points c0..c0+15 (mirrored A layout)
        v2f bf;
        {
            int c = c0 + sub;
            bf.x = lo ? xs[c] : zs[c];
            bf.y = lo ? ys[c] : 0.0f;
        }
        v8f cz = {};
        v8f g0 = __builtin_amdgcn_wmma_f32_16x16x4_f32(
            false, a0, false, bf, (short)0, cz, false, false);
        v8f g1 = __builtin_amdgcn_wmma_f32_16x16x4_f32(
            false, a1, false, bf, (short)0, cz, false, false);

        // transpose 32x16 Gram tile through LDS so each lane owns one row
#pragma unroll
        for (int v = 0; v < 8; ++v) {
            tbuf[(Mbase + v) * 17 + Ncol]      = g0[v];
            tbuf[(Mbase + v + 16) * 17 + Ncol] = g1[v];
        }
        // same-wave LDS ops are in-order; just fence compiler + DScnt
        asm volatile("s_wait_dscnt 0" ::: "memory");

#pragma unroll
        for (int j = 0; j < 16; ++j) {
            float G = tbuf[lane * 17 + j];
            float d = sqR + sqv[c0 + j] - 2.0f * G;
            if (d < nd[KNN - 1]) {
                nd[KNN - 1] = d; ni[KNN - 1] = c0 + j;
#pragma unroll
                for (int t = KNN - 2; t >= 0; --t) {
                    if (nd[t + 1] < nd[t]) {
                        float td = nd[t]; nd[t] = nd[t + 1]; nd[t + 1] = td;
                        int   ti = ni[t]; ni[t] = ni[t + 1]; ni[t + 1] = ti;
                    }
                }
            }
        }
        asm volatile("" ::: "memory");
    }

    // PCA normal of the 10-NN neighborhood (scatter matrix, no 1/k needed)
    float px[KNN], py[KNN], pz[KNN];
#pragma unroll
    for (int j = 0; j < KNN; ++j) {
        int n = ni[j];
        px[j] = xs[n]; py[j] = ys[n]; pz[j] = zs[n];
    }
    float mx = 0.0f, my = 0.0f, mz = 0.0f;
#pragma unroll
    for (int j = 0; j < KNN; ++j) { mx += px[j]; my += py[j]; mz += pz[j]; }
    mx *= (1.0f / KNN); my *= (1.0f / KNN); mz *= (1.0f / KNN);
    float cxx = 0, cxy = 0, cxz = 0, cyy = 0, cyz = 0, czz = 0;
#pragma unroll
    for (int j = 0; j < KNN; ++j) {
        float dx = px[j] - mx, dy = py[j] - my, dz = pz[j] - mz;
        cxx += dx * dx; cxy += dx * dy; cxz += dx * dz;
        cyy += dy * dy; cyz += dy * dz; czz += dz * dz;
    }
    float ex, ey, ez;
    smallest_evec(cxx, cxy, cxz, cyy, cyz, czz, ex, ey, ez);

    int row = rBase + lane;
    float* o = normals + ((size_t)b * NPTS + row) * 3;
    o[0] = ex; o[1] = ey; o[2] = ez;
}

// ---------------------------------------------------------------------------
// Kernel 3: loss = mean_b sum_n (1 - cos(pred_n, gt_n))
// ---------------------------------------------------------------------------
__global__ void loss_reduce(const float* __restrict__ gn,
                            const float* __restrict__ pn,
                            float* __restrict__ out) {
    __shared__ float red[256];
    const float scale = 1.0f / (float)BATCH;
    float acc = 0.0f;
    for (int i = blockIdx.x * blockDim.x + threadIdx.x; i < BATCH * NPTS;
         i += gridDim.x * blockDim.x) {
        float ax = gn[3 * i], ay = gn[3 * i + 1], az = gn[3 * i + 2];
        float bx = pn[3 * i], by = pn[3 * i + 1], bz = pn[3 * i + 2];
        float na = fmaxf(sqrtf(ax * ax + ay * ay + az * az), 1e-8f);
        float nb = fmaxf(sqrtf(bx * bx + by * by + bz * bz), 1e-8f);
        float c  = (ax * bx + ay * by + az * bz) / (na * nb);
        acc += (1.0f - c);
    }
    red[threadIdx.x] = acc * scale;
    __syncthreads();
    for (int s = 128; s > 0; s >>= 1) {
        if (threadIdx.x < (unsigned)s) red[threadIdx.x] += red[threadIdx.x + s];
        __syncthreads();
    }
    if (threadIdx.x == 0) atomicAdd(out, red[0]);
}

// ---------------------------------------------------------------------------
extern "C" void kernel_launch(void* const* d_in, const int* in_sizes, int n_in,
                              void* d_out, int out_size, void* d_ws, size_t ws_size,
                              hipStream_t stream) {
    (void)in_sizes; (void)n_in; (void)out_size; (void)ws_size;
    const float* gt    = (const float*)d_in[0];
    const float* pred  = (const float*)d_in[1];
    const int*   idx12 = (const int*)d_in[2];
    float* out = (float*)d_out;

    float* predg = (float*)d_ws;                      // BATCH*NPTS*3
    float* gn    = predg + (size_t)BATCH * NPTS * 3;  // gt normals
    float* pn    = gn    + (size_t)BATCH * NPTS * 3;  // pred normals

    gather_zero<<<(BATCH * NPTS + 255) / 256, 256, 0, stream>>>(pred, idx12, predg, out);

    dim3 grid(NPTS / 128, BATCH);
    knn_normals_kernel<<<grid, 128, 0, stream>>>(gt, gn);
    knn_normals_kernel<<<grid, 128, 0, stream>>>(predg, pn);

    loss_reduce<<<64, 256, 0, stream>>>(gn, pn, out);
}